// KeyValueMemoryNet_83202106458268
// MI455X (gfx1250) — compile-verified
//
#include <hip/hip_runtime.h>
#include <cstdint>

typedef float v2f __attribute__((ext_vector_type(2)));
typedef float v8f __attribute__((ext_vector_type(8)));

#define B_  256
#define S_  2048
#define I_  128
#define H_  32
#define K_  64
#define O_  256
#define M_  3
#define G3H 96          // 3*H
#define WPAD 130        // LDS row stride for W_ih staging (bank-conflict padding)

// =====================================================================
// Phase 1: gi[N x 96] = inputs[N x 128] @ W_ih^T[128 x 96] + b_ih
//          N = B*S = 524288.  One wave -> 16 rows x 96 cols,
//          fp32 WMMA 16x16x4 chained 32x over K=128, 6 column tiles.
// =====================================================================
__global__ void gi_gemm_kernel(const float* __restrict__ in,
                               const float* __restrict__ W_ih,
                               const float* __restrict__ b_ih,
                               float* __restrict__ gi) {
  extern __shared__ float wlds[];              // [96][WPAD]
  const int tid = threadIdx.x;
  for (int i = tid; i < G3H * I_; i += blockDim.x) {
    int r = i >> 7;                            // W_ih row (output col)
    int k = i & (I_ - 1);                      // input dim
    wlds[r * WPAD + k] = W_ih[i];
  }
  __syncthreads();

  const int wave = tid >> 5;
  const int lane = tid & 31;
  const int m16  = lane & 15;
  const int hi   = lane >> 4;                  // lane half selects K pair / M half

  const long rowTile = (long)blockIdx.x * 8 + wave;
  const long r0 = rowTile * 16;

  v8f acc[6];
  #pragma unroll
  for (int n = 0; n < 6; ++n) acc[n] = (v8f){0.f,0.f,0.f,0.f,0.f,0.f,0.f,0.f};

  // A fragment base: 16x4 f32 layout -> lanes 0-15 hold K={0,1}, 16-31 K={2,3}
  const float* aBase = in + (r0 + m16) * I_ + 2 * hi;

  for (int c = 0; c < 32; ++c) {               // K chunks of 4
    v2f a = *(const v2f*)(aBase + 4 * c);
    #pragma unroll
    for (int n = 0; n < 6; ++n) {
      // B fragment: 4x16 f32, B[k][col] = W_ih[col][k]
      v2f bf = *(const v2f*)(&wlds[(n * 16 + m16) * WPAD + 4 * c + 2 * hi]);
      acc[n] = __builtin_amdgcn_wmma_f32_16x16x4_f32(
          /*neg_a=*/false, a, /*neg_b=*/false, bf,
          /*c_mod=*/(short)0, acc[n], /*reuse_a=*/false, /*reuse_b=*/false);
    }
  }

  // D layout: VGPR i -> M = i + 8*hi, N = m16
  #pragma unroll
  for (int n = 0; n < 6; ++n) {
    float bias = b_ih[n * 16 + m16];
    #pragma unroll
    for (int i = 0; i < 8; ++i) {
      long row = r0 + i + 8 * hi;
      gi[row * G3H + n * 16 + m16] = acc[n][i] + bias;
    }
  }
}

// =====================================================================
// Phase 2: sequential GRU + KV-memory scan. One wave per batch row,
// lane j owns hidden unit j (H==32==wave32). W_hh/Wq rows in VGPRs,
// h broadcast via shuffles, KV memory in LDS.
// =====================================================================
template <bool FUSED>
__global__ void scan_kernel(const float* __restrict__ gi_all,
                            const float* __restrict__ in,
                            const float* __restrict__ W_ih,
                            const float* __restrict__ b_ih,
                            const float* __restrict__ W_hh,
                            const float* __restrict__ b_hh,
                            const float* __restrict__ Wk,
                            const float* __restrict__ bk,
                            const float* __restrict__ Wq,
                            const float* __restrict__ bq,
                            const float* __restrict__ Wv,
                            const float* __restrict__ bv,
                            float* __restrict__ outp,
                            float* __restrict__ hidp) {
  extern __shared__ float smem[];
  const int tid  = threadIdx.x;
  const int wave = tid >> 5;
  const int j    = tid & 31;

  float* wihT = smem;                                           // [128][96] (FUSED only)
  float* kv   = smem + (FUSED ? I_ * G3H : 0) + wave * (M_ * K_ + M_ * O_);
  float* keys = kv;                                             // [3][64]
  float* vals = kv + M_ * K_;                                   // [3][256]

  if (FUSED) {
    for (int i = tid; i < I_ * G3H; i += blockDim.x) {
      int r = i >> 7;                  // W_ih row
      int k = i & (I_ - 1);
      wihT[k * G3H + r] = W_ih[i];     // transposed -> conflict-free reads
    }
  }
  for (int i = j; i < M_ * K_ + M_ * O_; i += 32) kv[i] = 0.f;  // values start at 0
  if (FUSED) __syncthreads();

  const int b = blockIdx.x * 8 + wave;

  // per-lane weight rows kept in VGPRs
  float wr[H_], wz[H_], wn[H_], wq0[H_], wq1[H_];
  #pragma unroll
  for (int k = 0; k < H_; ++k) {
    wr[k]  = W_hh[(0 * H_ + j) * H_ + k];
    wz[k]  = W_hh[(1 * H_ + j) * H_ + k];
    wn[k]  = W_hh[(2 * H_ + j) * H_ + k];
    wq0[k] = Wq[j * H_ + k];
    wq1[k] = Wq[(H_ + j) * H_ + k];
  }
  const float bhr = b_hh[j], bhz = b_hh[H_ + j], bhn = b_hh[2 * H_ + j];
  const float bq0 = bq[j],   bq1 = bq[H_ + j];
  float bir = 0.f, biz = 0.f, bin_ = 0.f;
  if (FUSED) { bir = b_ih[j]; biz = b_ih[H_ + j]; bin_ = b_ih[2 * H_ + j]; }

  float h = 0.f;

  for (int t = 0; t < S_; ++t) {
    // ---- gi (precomputed by WMMA GEMM, or fused fallback)
    float gir, giz, gin;
    if (FUSED) {
      const float* x = in + ((long)b * S_ + t) * I_;
      float xr[4];
      #pragma unroll
      for (int q = 0; q < 4; ++q) xr[q] = x[q * 32 + j];
      gir = bir; giz = biz; gin = bin_;
      #pragma unroll
      for (int k = 0; k < I_; ++k) {
        float xk = __shfl(xr[k >> 5], k & 31, 32);
        const float* wrow = &wihT[k * G3H];
        gir += xk * wrow[j];
        giz += xk * wrow[H_ + j];
        gin += xk * wrow[2 * H_ + j];
      }
    } else {
      const float* g = gi_all + ((long)b * S_ + t) * G3H;
      gir = g[j]; giz = g[H_ + j]; gin = g[2 * H_ + j];
    }

    // ---- gh = h @ W_hh^T + b_hh (broadcast old h across lanes)
    float ghr = bhr, ghz = bhz, ghn = bhn;
    #pragma unroll
    for (int k = 0; k < H_; ++k) {
      float hk = __shfl(h, k, 32);
      ghr += hk * wr[k];
      ghz += hk * wz[k];
      ghn += hk * wn[k];
    }
    float r  = 1.f / (1.f + __expf(-(gir + ghr)));
    float z  = 1.f / (1.f + __expf(-(giz + ghz)));
    float nx = gin + r * ghn;
    float n  = 2.f / (1.f + __expf(-2.f * nx)) - 1.f;   // tanh
    h = (1.f - z) * n + z * h;

    hidp[((long)b * S_ + t) * H_ + j] = h;

    // ---- q = h @ Wq^T + bq (new h)
    float q0 = bq0, q1 = bq1;
    #pragma unroll
    for (int k = 0; k < H_; ++k) {
      float hk = __shfl(h, k, 32);
      q0 += hk * wq0[k];
      q1 += hk * wq1[k];
    }

    // ---- scores over M=3 slots (mask m < t), softmax
    float s[M_];
    #pragma unroll
    for (int m = 0; m < M_; ++m) {
      float p = q0 * keys[m * K_ + j] + q1 * keys[m * K_ + H_ + j];
      #pragma unroll
      for (int off = 16; off > 0; off >>= 1) p += __shfl_xor(p, off, 32);
      s[m] = (m < t) ? p : -1e30f;
    }
    float mx  = fmaxf(s[0], fmaxf(s[1], s[2]));
    float e0  = __expf(s[0] - mx), e1 = __expf(s[1] - mx), e2 = __expf(s[2] - mx);
    float inv = 1.f / (e0 + e1 + e2);
    float w0 = e0 * inv, w1 = e1 * inv, w2 = e2 * inv;
    float gate = (t > 0) ? 1.f : 0.f;

    // ---- retrieved = w @ values  (8 coalesced outputs per lane)
    float* ob = outp + ((long)b * S_ + t) * O_;
    #pragma unroll
    for (int i = 0; i < 8; ++i) {
      int o = i * 32 + j;
      float rv = w0 * vals[0 * O_ + o] + w1 * vals[1 * O_ + o] + w2 * vals[2 * O_ + o];
      ob[o] = rv * gate;
    }

    // ---- write slot t (only first M steps; wave-uniform branch)
    if (t < M_) {
      float kk0 = bk[j], kk1 = bk[H_ + j];
      float va[8];
      #pragma unroll
      for (int i = 0; i < 8; ++i) va[i] = bv[i * 32 + j];
      #pragma unroll
      for (int k = 0; k < H_; ++k) {
        float hk = __shfl(h, k, 32);
        kk0 += hk * Wk[j * H_ + k];
        kk1 += hk * Wk[(H_ + j) * H_ + k];
        #pragma unroll
        for (int i = 0; i < 8; ++i) va[i] += hk * Wv[(i * 32 + j) * H_ + k];
      }
      keys[t * K_ + j]      = kk0;
      keys[t * K_ + H_ + j] = kk1;
      #pragma unroll
      for (int i = 0; i < 8; ++i) vals[t * O_ + i * 32 + j] = va[i];
    }
  }
}

// =====================================================================
extern "C" void kernel_launch(void* const* d_in, const int* in_sizes, int n_in,
                              void* d_out, int out_size, void* d_ws, size_t ws_size,
                              hipStream_t stream) {
  const float* inputs = (const float*)d_in[0];
  const float* W_ih   = (const float*)d_in[1];
  const float* W_hh   = (const float*)d_in[2];
  const float* b_ih   = (const float*)d_in[3];
  const float* b_hh   = (const float*)d_in[4];
  const float* Wk     = (const float*)d_in[5];
  const float* bk     = (const float*)d_in[6];
  const float* Wq     = (const float*)d_in[7];
  const float* bq     = (const float*)d_in[8];
  const float* Wv     = (const float*)d_in[9];
  const float* bv     = (const float*)d_in[10];

  float* outp = (float*)d_out;
  float* hidp = outp + (size_t)B_ * S_ * O_;

  const size_t giBytes = (size_t)B_ * S_ * G3H * sizeof(float);
  const size_t kvBytes = (size_t)8 * (M_ * K_ + M_ * O_) * sizeof(float);

  if (ws_size >= giBytes) {
    float* gi = (float*)d_ws;
    const int rowTiles = (B_ * S_) / 16;    // 32768 row tiles, 8 waves/WG
    gi_gemm_kernel<<<rowTiles / 8, 256, G3H * WPAD * sizeof(float), stream>>>(
        inputs, W_ih, b_ih, gi);
    scan_kernel<false><<<B_ / 8, 256, kvBytes, stream>>>(
        gi, inputs, W_ih, b_ih, W_hh, b_hh, Wk, bk, Wq, bq, Wv, bv, outp, hidp);
  } else {
    scan_kernel<true><<<B_ / 8, 256, (size_t)I_ * G3H * sizeof(float) + kvBytes, stream>>>(
        nullptr, inputs, W_ih, b_ih, W_hh, b_hh, Wk, bk, Wq, bq, Wv, bv, outp, hidp);
  }
}